// SparseScatter_63488206569807
// MI455X (gfx1250) — compile-verified
//
#include <hip/hip_runtime.h>
#include <stdint.h>

// ---------------------------------------------------------------------------
// SparseScatter: y = y_base; y[n, yb*16:+16, xb*16:+16, :] = transpose(x[i])
//   x:      (4096, 32, 16, 16) f32   (channel-major blocks)
//   y_base: (8, 512, 512, 32)  f32   (NHWC)
//   indices:(4096, 3) i32            (n, yb, xb), distinct blocks
// Pure data movement; optimized for HBM bandwidth on MI455X:
//   - one write pass over y (mask of active blocks in d_ws)
//   - CDNA5 async global->LDS (b128) for the per-block C-major -> HWC transpose
//   - non-temporal hints for all streaming global traffic (y >> L2)
// ---------------------------------------------------------------------------

namespace {
constexpr int kB  = 8;
constexpr int kH  = 512;
constexpr int kW  = 512;
constexpr int kC  = 32;
constexpr int kBH = 16;
constexpr int kBW = 16;
constexpr int kGH = kH / kBH;                    // 32
constexpr int kGW = kW / kBW;                    // 32
constexpr int kNBlkTotal = kB * kGH * kGW;       // 8192
constexpr int kNActive   = 4096;
constexpr int kBlkFloats = kC * kBH * kBW;       // 8192 floats = 32 KB
constexpr int kLdsStride = 260;                  // floats per channel row (16B-aligned pad)
}

typedef __attribute__((ext_vector_type(4))) float v4f;

__global__ __launch_bounds__(256) void init_mask_kernel(int* __restrict__ mask) {
  int i = blockIdx.x * 256 + threadIdx.x;
  if (i < kNBlkTotal) mask[i] = -1;
}

__global__ __launch_bounds__(256) void set_mask_kernel(const int* __restrict__ idx,
                                                       int* __restrict__ mask) {
  int i = blockIdx.x * 256 + threadIdx.x;
  if (i < kNActive) {
    int n  = idx[3 * i + 0];
    int by = idx[3 * i + 1];
    int bx = idx[3 * i + 2];
    mask[(n * kGH + by) * kGW + bx] = i;
  }
}

// Stage one x-block (C-major) into LDS via async global->LDS b128 with
// scattered per-lane LDS addresses, then emit it HWC with coalesced NT stores.
__device__ __forceinline__ void transpose_scatter(const float* __restrict__ x,
                                                  float* __restrict__ out,
                                                  int act, size_t base, int t,
                                                  float* lds) {
  const char* src = (const char*)(x + (size_t)act * kBlkFloats);
  uint32_t lbase = (uint32_t)(uintptr_t)lds;   // low 32 bits of generic ptr = LDS byte offset
  #pragma unroll
  for (int k = 0; k < 8; ++k) {
    int f  = t + k * 256;                      // b128 index within block (0..2047)
    int ch = f >> 6;                           // 64 x b128 per channel
    int p  = (f & 63) << 2;                    // first pixel of this b128
    uint32_t laddr = lbase + (uint32_t)(ch * kLdsStride + p) * 4u;
    uint64_t gaddr = (uint64_t)(uintptr_t)(src + (size_t)f * 16u);
    asm volatile("global_load_async_to_lds_b128 %0, %1, off"
                 :: "v"(laddr), "v"(gaddr)
                 : "memory");
  }
  asm volatile("s_wait_asynccnt 0" ::: "memory");
  __syncthreads();
  #pragma unroll
  for (int k = 0; k < 8; ++k) {
    int o   = t + k * 256;                     // output float4 index (0..2047)
    int pix = o >> 3;                          // pixel 0..255
    int c4  = (o & 7) << 2;                    // starting channel
    v4f v;
    v.x = lds[(c4 + 0) * kLdsStride + pix];
    v.y = lds[(c4 + 1) * kLdsStride + pix];
    v.z = lds[(c4 + 2) * kLdsStride + pix];
    v.w = lds[(c4 + 3) * kLdsStride + pix];
    int row = pix >> 4, col = pix & 15;
    size_t off = base + (size_t)row * (kW * kC) + (size_t)(col * kC + c4);
    __builtin_nontemporal_store(v, (v4f*)(out + off));   // streaming write, TH=NT
  }
}

// One workgroup per grid block (8192 total): copy-or-scatter exactly once.
__global__ __launch_bounds__(256) void combined_kernel(const float* __restrict__ x,
                                                       const float* __restrict__ ybase,
                                                       const int* __restrict__ mask,
                                                       float* __restrict__ out) {
  __shared__ float lds[kC * kLdsStride];
  int g = blockIdx.x;
  int t = threadIdx.x;
  int n  = g >> 10;
  int by = (g >> 5) & 31;
  int bx = g & 31;
  size_t base = (((size_t)n * kH + (size_t)by * kBH) * kW + (size_t)bx * kBW) * kC;
  int act = mask[g];
  if (act >= 0) {
    transpose_scatter(x, out, act, base, t, lds);   // uniform branch per workgroup
  } else {
    const v4f* s4 = (const v4f*)ybase;
    v4f* d4 = (v4f*)out;
    #pragma unroll
    for (int k = 0; k < 8; ++k) {
      int v    = t + k * 256;                 // float4 index within block (0..2047)
      int row  = v >> 7;                      // 128 float4 per 16-pixel row
      int col4 = v & 127;
      size_t off4 = (base + (size_t)row * (kW * kC)) / 4 + col4;
      v4f val = __builtin_nontemporal_load(s4 + off4);  // read-once, TH=NT
      __builtin_nontemporal_store(val, d4 + off4);
    }
  }
}

// ---- fallback path (workspace too small for the mask) ----------------------
__global__ __launch_bounds__(256) void copy_all_kernel(const v4f* __restrict__ src,
                                                       v4f* __restrict__ dst, size_t n4) {
  size_t i = (size_t)blockIdx.x * 256 + threadIdx.x;
  size_t stride = (size_t)gridDim.x * 256;
  for (; i < n4; i += stride) {
    v4f val = __builtin_nontemporal_load(src + i);
    __builtin_nontemporal_store(val, dst + i);
  }
}

__global__ __launch_bounds__(256) void scatter_kernel(const float* __restrict__ x,
                                                      const int* __restrict__ idx,
                                                      float* __restrict__ out) {
  __shared__ float lds[kC * kLdsStride];
  int i = blockIdx.x;
  int t = threadIdx.x;
  int n  = idx[3 * i + 0];
  int by = idx[3 * i + 1];
  int bx = idx[3 * i + 2];
  size_t base = (((size_t)n * kH + (size_t)by * kBH) * kW + (size_t)bx * kBW) * kC;
  transpose_scatter(x, out, i, base, t, lds);
}

extern "C" void kernel_launch(void* const* d_in, const int* in_sizes, int n_in,
                              void* d_out, int out_size, void* d_ws, size_t ws_size,
                              hipStream_t stream) {
  const float* x     = (const float*)d_in[0];
  const float* ybase = (const float*)d_in[1];
  const int*   idx   = (const int*)d_in[2];
  float* out = (float*)d_out;

  if (ws_size >= (size_t)kNBlkTotal * sizeof(int)) {
    int* mask = (int*)d_ws;
    init_mask_kernel<<<kNBlkTotal / 256, 256, 0, stream>>>(mask);
    set_mask_kernel<<<kNActive / 256, 256, 0, stream>>>(idx, mask);
    combined_kernel<<<kNBlkTotal, 256, 0, stream>>>(x, ybase, mask, out);
  } else {
    size_t n4 = (size_t)kB * kH * kW * kC / 4;
    copy_all_kernel<<<4096, 256, 0, stream>>>((const v4f*)ybase, (v4f*)out, n4);
    scatter_kernel<<<kNActive, 256, 0, stream>>>(x, idx, out);
  }
}